// Symmetrizer_Vectorized_2843268350084
// MI455X (gfx1250) — compile-verified
//
#include <hip/hip_runtime.h>
#include <hip/hip_bf16.h>

typedef float v2f __attribute__((ext_vector_type(2)));
typedef float v8f __attribute__((ext_vector_type(8)));

// out(4x16) = P(4x20) * g(20x16) per (node,radial) pair, where
//   g[0][c] = x[0][c], g[k][c] = x[k][c]^2 (k>=1)
// realized as 5 chained V_WMMA_F32_16X16X4_F32 (K=20 in chunks of 4).
__global__ __launch_bounds__(256) void symmetrizer_wmma_kernel(
    const float* __restrict__ x, float* __restrict__ out, int npairs)
{
    const int lane = threadIdx.x & 31;
    const int lo   = lane & 15;   // M (slot row) for A; N (channel) for B/D
    const int hi   = lane >> 4;   // K-half selector within a 4-wide chunk

    const int wavesPerBlock = blockDim.x >> 5;
    const int gwave  = blockIdx.x * wavesPerBlock + (threadIdx.x >> 5);
    const int nwaves = gridDim.x * wavesPerBlock;

    // Hardcoded tables from _build_tables(MAX_L=3):
    // slot-of-k (k=0 is the passthrough row -> slot 0), multinomial prefactors.
    const int   slotmap[20] = {0, 1,1,1, 2,2,2,2,2,2, 3,3,3,3,3,3,3,3,3,3};
    const float prefs[20]   = {1.f, 1.f,1.f,1.f,
                               1.f,2.f,2.f,1.f,2.f,1.f,
                               1.f,3.f,3.f,3.f,6.f,3.f,1.f,3.f,3.f,1.f};

    // Constant A-operands (P chunks), built once per thread.
    // A 16x4 fp32 layout: lane L holds M=L%16; V0 = K(0+2*hi), V1 = K(1+2*hi).
    v2f aop[5];
#pragma unroll
    for (int c = 0; c < 5; ++c) {
        const int kA = 4 * c + 2 * hi;
        aop[c].x = (lo == slotmap[kA])     ? prefs[kA]     : 0.0f;
        aop[c].y = (lo == slotmap[kA + 1]) ? prefs[kA + 1] : 0.0f;
    }

    // B 4x16 fp32 layout: lane L holds N=L%16; V0 = row K(0+2*hi), V1 = K(1+2*hi).
    // Element (k,c) of the tile sits at offset k*16 + c, so:
    const int off = lo + 32 * hi;   // = (2*hi)*16 + lo

    for (int pair = gwave; pair < npairs; pair += nwaves) {
        const float* base = x + (size_t)pair * 320;   // 20 ang * 16 ch

        v8f acc = {0.f, 0.f, 0.f, 0.f, 0.f, 0.f, 0.f, 0.f};
#pragma unroll
        for (int c = 0; c < 5; ++c) {
            // Two b32 loads per chunk; each instruction covers two 64-byte runs, 128 bytes apart.
            float v0 = base[64 * c + off];
            float v1 = base[64 * c + off + 16];
            float b0 = v0 * v0;
            float b1 = v1 * v1;
            if (c == 0) {
                // k==0 row (lanes with hi==0 in V0 of chunk 0) passes through un-squared.
                b0 = (hi != 0) ? b0 : v0;
            }
            v2f b;
            b.x = b0;
            b.y = b1;
            acc = __builtin_amdgcn_wmma_f32_16x16x4_f32(
                /*neg_a=*/false, aop[c], /*neg_b=*/false, b,
                /*c_mod=*/(short)0, acc, /*reuse_a=*/false, /*reuse_b=*/false);
        }

        // D rows 0..3 (slots) live in acc[0..3] of lanes 0..15 (N = lane).
        float* o = out + (size_t)pair * 64;           // 4 slots * 16 ch
        if (lane < 16) {
            o[lo]      = acc[0];
            o[16 + lo] = acc[1];
            o[32 + lo] = acc[2];
            o[48 + lo] = acc[3];
        }
    }
}

extern "C" void kernel_launch(void* const* d_in, const int* in_sizes, int n_in,
                              void* d_out, int out_size, void* d_ws, size_t ws_size,
                              hipStream_t stream) {
    const float* node_attr = (const float*)d_in[0];
    float* out = (float*)d_out;

    // in_sizes[0] = N * R * 20 * 16 ; one "pair" = one 20x16 tile.
    const int npairs = in_sizes[0] / 320;

    const int threads = 256;                 // 8 waves per block
    int blocks = 2048;                       // ~16K waves: enough to saturate HBM
    const int blocksNeeded = (npairs + 7) / 8;
    if (blocks > blocksNeeded) blocks = blocksNeeded;
    if (blocks < 1) blocks = 1;

    symmetrizer_wmma_kernel<<<blocks, threads, 0, stream>>>(node_attr, out, npairs);
}